// decoder_block_67035849556225
// MI455X (gfx1250) — compile-verified
//
#include <hip/hip_runtime.h>
#include <hip/hip_bf16.h>
#include <math.h>

#define DEVI __device__ __forceinline__

typedef __attribute__((ext_vector_type(2))) float v2f;
typedef __attribute__((ext_vector_type(8))) float v8f;
typedef __attribute__((ext_vector_type(4))) unsigned int u32x4;
typedef __attribute__((ext_vector_type(4))) int i32x4;
typedef __attribute__((ext_vector_type(8))) int i32x8;

// ---------------- problem dims ----------------
constexpr int B = 16, INC = 512, OUTC = 256;
constexpr int FD = 1024;          // 32*32
constexpr int DK = 128;           // INC/4
constexpr int H2 = 64, W2 = 64, P2 = H2 * W2;   // upsampled plane = 4096
constexpr int NT = B - 2;         // 14 interior triplets
constexpr int IC2 = 2 * OUTC;     // conv3 input channels = 512

// ---------------- workspace layout (floats) ----------------
constexpr size_t OFF_NEW = 0;                                   // B*INC*FD     post-attn activations
constexpr size_t OFF_Q   = OFF_NEW + (size_t)B * INC * FD;      // NT*INC*DK
constexpr size_t OFF_KL  = OFF_Q   + (size_t)NT * INC * DK;
constexpr size_t OFF_KR  = OFF_KL  + (size_t)NT * INC * DK;
constexpr size_t OFF_KV  = OFF_KR  + (size_t)NT * INC * DK;     // NT*2*DK*FD
constexpr size_t OFF_KS  = OFF_KV  + (size_t)NT * 2 * DK * FD;  // NT*2*DK
constexpr size_t OFF_Z   = OFF_KS  + (size_t)NT * 2 * DK;       // NT*2*INC
constexpr size_t OFF_G   = OFF_Z   + (size_t)NT * 2 * INC;      // B*OUTC*P2    gated branch
constexpr size_t OFF_Y   = OFF_G   + (size_t)B * OUTC * P2;     // B*OUTC*P2    pre-BN conv out
constexpr size_t OFF_ST  = OFF_Y   + (size_t)B * OUTC * P2;     // 2*OUTC       mu, rstd
constexpr size_t OFF_W2  = OFF_ST  + (size_t)2 * OUTC;          // 9*IC2*OUTC   conv weights [tap][ic][oc]

// ---------------- WMMA micro-kernel: wave owns a (16*MR)x(16*NR) f32 block ----------------
DEVI v8f vzero() { v8f z = {0.f,0.f,0.f,0.f,0.f,0.f,0.f,0.f}; return z; }

DEVI v8f wmma4(v2f a, v2f b, v8f c) {
  // D = A(16x4) * B(4x16) + C, full fp32  ->  v_wmma_f32_16x16x4_f32
  return __builtin_amdgcn_wmma_f32_16x16x4_f32(false, a, false, b, (short)0, c,
                                               false, false);
}

// lda(m,k): A[(16*MR) x K], ldb(k,n): B[K x (16*NR)]
template <int MR, int NR, typename FA, typename FB>
DEVI void gemm_tile(int K, FA&& lda, FB&& ldb, v8f (&acc)[MR][NR]) {
  const int lane = threadIdx.x & 31;
  const int half = lane >> 4;      // K pair select
  const int idx  = lane & 15;
#pragma unroll 2
  for (int k = 0; k < K; k += 4) {
    const int ka = k + 2 * half;
    v2f a[MR], b[NR];
#pragma unroll
    for (int i = 0; i < MR; ++i) {
      a[i].x = lda(16 * i + idx, ka);
      a[i].y = lda(16 * i + idx, ka + 1);
    }
#pragma unroll
    for (int j = 0; j < NR; ++j) {
      b[j].x = ldb(ka, 16 * j + idx);
      b[j].y = ldb(ka + 1, 16 * j + idx);
    }
#pragma unroll
    for (int i = 0; i < MR; ++i)
#pragma unroll
      for (int j = 0; j < NR; ++j) acc[i][j] = wmma4(a[i], b[j], acc[i][j]);
  }
}

// st(row, col, v) over the (16*MR)x(16*NR) block
template <int MR, int NR, typename FS>
DEVI void store_block(const v8f (&acc)[MR][NR], FS&& st) {
  const int lane = threadIdx.x & 31;
  const int half = lane >> 4;
  const int col  = lane & 15;
#pragma unroll
  for (int i = 0; i < MR; ++i)
#pragma unroll
    for (int j = 0; j < NR; ++j)
#pragma unroll
      for (int r = 0; r < 8; ++r)
        st(16 * i + r + 8 * half, 16 * j + col, acc[i][j][r]);
}

DEVI float phi_elu1(float v) { return v > 0.f ? v + 1.f : __expf(v); }

// ---------------- TDM: 2D f32 tile -> LDS (D# per CDNA5 ISA) ----------------
// tensor: rows of tensor_d0 elements, row stride stride_d0 (elements), tensor_d1 rows
// tile:   tile_d0 x tile_d1 elements starting at gaddr, packed row-major into lds_addr
DEVI void tdm_load_2d_f32(unsigned lds_addr, const void* gaddr,
                          unsigned tensor_d0, unsigned tensor_d1, unsigned stride_d0,
                          unsigned tile_d0, unsigned tile_d1) {
  const unsigned long long ga = (unsigned long long)(uintptr_t)gaddr;
  u32x4 g0;
  g0[0] = 1u;                                            // count=1 valid descriptor
  g0[1] = lds_addr;                                      // LDS byte address
  g0[2] = (unsigned)ga;                                  // global_addr[31:0]
  g0[3] = (unsigned)((ga >> 32) & 0x01FFFFFFu) | (2u << 30);  // addr[56:32] | type=2
  i32x8 g1;
  g1[0] = (int)(2u << 16);                               // data_size = 4 bytes
  g1[1] = (int)(tensor_d0 << 16);                        // tensor_dim0[15:0]
  g1[2] = (int)((tensor_d0 >> 16) | (tensor_d1 << 16));  // dim0[31:16] | dim1[15:0]
  g1[3] = (int)((tensor_d1 >> 16) | (tile_d0 << 16));    // dim1[31:16] | tile_dim0
  g1[4] = (int)tile_d1;                                  // tile_dim1 (tile_dim2 = 0)
  g1[5] = (int)stride_d0;                                // tensor_dim0_stride[31:0]
  g1[6] = 0;                                             // stride0[47:32] | stride1[15:0]
  g1[7] = 0;
  i32x4 z4 = {0, 0, 0, 0};
#if defined(__clang_major__) && (__clang_major__ >= 23)
  i32x8 z8 = {0, 0, 0, 0, 0, 0, 0, 0};
  __builtin_amdgcn_tensor_load_to_lds(g0, g1, z4, z4, z8, 0);
#else
  __builtin_amdgcn_tensor_load_to_lds(g0, g1, z4, z4, 0);
#endif
}

// ---------------- K0: pass-through samples 0 and B-1 ----------------
__global__ void k_copy_edges(const float* __restrict__ x, float* __restrict__ ws) {
  constexpr size_t n4 = (size_t)INC * FD / 4;
  size_t i = (size_t)blockIdx.x * blockDim.x + threadIdx.x;
  if (i >= n4) return;
  const float4* xs = (const float4*)x;
  float4* d = (float4*)(ws + OFF_NEW);
  d[i] = xs[i];
  d[(size_t)(B - 1) * n4 + i] = xs[(size_t)(B - 1) * n4 + i];
}

// ---------------- K1: projections phi(slice @ W), 42 GEMMs (M=512,K=1024,N=128) -----
__global__ void k_proj(const float* __restrict__ x,
                       const float* __restrict__ wq,
                       const float* __restrict__ wkl,
                       const float* __restrict__ wkr,
                       float* __restrict__ ws) {
  const int wave = (int)((blockIdx.x * blockDim.x + threadIdx.x) >> 5);
  constexpr int TN = DK / 32, TILES = (INC / 32) * TN;   // 16*4
  const int g = wave / TILES;
  if (g >= 3 * NT) return;
  const int tile = wave - g * TILES;
  const int tm = tile / TN, tn = tile - tm * TN;
  const int t = g / 3, which = g - 3 * t;
  const float* src; const float* w; float* dst;
  if (which == 0)      { src = x + (size_t)(t + 1) * INC * FD; w = wq;  dst = ws + OFF_Q  + (size_t)t * INC * DK; }
  else if (which == 1) { src = x + (size_t)(t    ) * INC * FD; w = wkl; dst = ws + OFF_KL + (size_t)t * INC * DK; }
  else                 { src = x + (size_t)(t + 2) * INC * FD; w = wkr; dst = ws + OFF_KR + (size_t)t * INC * DK; }
  const int m0 = tm * 32, n0 = tn * 32;
  v8f acc[2][2] = {{vzero(), vzero()}, {vzero(), vzero()}};
  gemm_tile<2, 2>(FD,
      [&](int m, int k) { return src[(size_t)(m0 + m) * FD + k]; },
      [&](int k, int n) { return w[(size_t)k * DK + n0 + n]; }, acc);
  store_block<2, 2>(acc, [&](int r, int c, float v) {
      dst[(size_t)(m0 + r) * DK + n0 + c] = phi_elu1(v);
  });
}

// ---------------- K2: ksum[t,side,dk] = sum_n K[n,dk] ----------------
__global__ void k_ksum(float* __restrict__ ws) {
  const int i = blockIdx.x * blockDim.x + threadIdx.x;
  if (i >= NT * 2 * DK) return;
  const int dk = i & (DK - 1);
  const int ts = i >> 7;                 // t*2+side
  const int side = ts & 1, t = ts >> 1;
  const float* K = ws + (side == 0 ? OFF_KL : OFF_KR) + (size_t)t * INC * DK;
  float s = 0.f;
  for (int n = 0; n < INC; ++n) s += K[(size_t)n * DK + dk];
  ws[OFF_KS + i] = s;
}

// ---------------- K3: kv = K^T @ V, 28 GEMMs (M=128,K=512,N=1024) ----------------
__global__ void k_kv(const float* __restrict__ x, float* __restrict__ ws) {
  const int wave = (int)((blockIdx.x * blockDim.x + threadIdx.x) >> 5);
  constexpr int TN = FD / 32, TILES = (DK / 32) * TN;    // 4*32
  const int g = wave / TILES;
  if (g >= NT * 2) return;
  const int tile = wave - g * TILES;
  const int tm = tile / TN, tn = tile - tm * TN;
  const int side = g & 1, t = g >> 1;
  const float* Kmat = ws + (side == 0 ? OFF_KL : OFF_KR) + (size_t)t * INC * DK;
  const float* V = x + (size_t)(side == 0 ? t : t + 2) * INC * FD;
  float* dst = ws + OFF_KV + (size_t)g * DK * FD;
  const int m0 = tm * 32, n0 = tn * 32;
  v8f acc[2][2] = {{vzero(), vzero()}, {vzero(), vzero()}};
  gemm_tile<2, 2>(INC,
      [&](int m, int k) { return Kmat[(size_t)k * DK + m0 + m]; },   // A = K^T
      [&](int k, int n) { return V[(size_t)k * FD + n0 + n]; }, acc);
  store_block<2, 2>(acc, [&](int r, int c, float v) {
      dst[(size_t)(m0 + r) * FD + n0 + c] = v;
  });
}

// ---------------- K4: z[t,side,m] = q[m]·ksum + 1e-6 ----------------
__global__ void k_z(float* __restrict__ ws) {
  const int i = blockIdx.x * blockDim.x + threadIdx.x;
  if (i >= NT * 2 * INC) return;
  const int m = i & (INC - 1);
  const int ts = i >> 9;
  const int t = ts >> 1;
  const float* Q = ws + OFF_Q + (size_t)t * INC * DK + (size_t)m * DK;
  const float* ksum = ws + OFF_KS + (size_t)ts * DK;
  float s = 1e-6f;
  for (int d = 0; d < DK; ++d) s += Q[d] * ksum[d];
  ws[OFF_Z + i] = s;
}

// ---------------- K5: out = x + (Q@kv_l)/z_l + (Q@kv_r)/z_r  (M=512,K=128,N=1024) ---
__global__ void k_attn(const float* __restrict__ x, float* __restrict__ ws) {
  const int wave = (int)((blockIdx.x * blockDim.x + threadIdx.x) >> 5);
  constexpr int TN = FD / 32, TILES = (INC / 32) * TN;   // 16*32
  const int t = wave / TILES;
  if (t >= NT) return;
  const int tile = wave - t * TILES;
  const int tm = tile / TN, tn = tile - tm * TN;
  const int m0 = tm * 32, n0 = tn * 32;
  const float* Q   = ws + OFF_Q  + (size_t)t * INC * DK;
  const float* kvl = ws + OFF_KV + (size_t)(t * 2 + 0) * DK * FD;
  const float* kvr = ws + OFF_KV + (size_t)(t * 2 + 1) * DK * FD;
  const float* xc  = x + (size_t)(t + 1) * INC * FD;
  const float* zl  = ws + OFF_Z + (size_t)(t * 2 + 0) * INC;
  const float* zr  = ws + OFF_Z + (size_t)(t * 2 + 1) * INC;
  float* dst = ws + OFF_NEW + (size_t)(t + 1) * INC * FD;
  auto lda = [&](int m, int k) { return Q[(size_t)(m0 + m) * DK + k]; };
  v8f accl[2][2] = {{vzero(), vzero()}, {vzero(), vzero()}};
  v8f accr[2][2] = {{vzero(), vzero()}, {vzero(), vzero()}};
  gemm_tile<2, 2>(DK, lda, [&](int k, int n) { return kvl[(size_t)k * FD + n0 + n]; }, accl);
  gemm_tile<2, 2>(DK, lda, [&](int k, int n) { return kvr[(size_t)k * FD + n0 + n]; }, accr);
  const int lane = threadIdx.x & 31;
  const int half = lane >> 4, col = lane & 15;
#pragma unroll
  for (int i = 0; i < 2; ++i)
#pragma unroll
    for (int j = 0; j < 2; ++j)
#pragma unroll
      for (int r = 0; r < 8; ++r) {
        const int m = m0 + 16 * i + r + 8 * half;
        const int n = n0 + 16 * j + col;
        const float v = xc[(size_t)m * FD + n] + accl[i][j][r] / zl[m] + accr[i][j][r] / zr[m];
        dst[(size_t)m * FD + n] = v;
      }
}

// ---------------- K6: ConvTranspose(k2,s2) GEMM + fused sigmoid(relu(up+s)) ---------
// M = B*FD (pixel-major), N = OUTC*4, K = INC
__global__ void k_upgate(const float* __restrict__ s,
                         const float* __restrict__ up_w,
                         const float* __restrict__ up_b,
                         float* __restrict__ ws) {
  const int wave = (int)((blockIdx.x * blockDim.x + threadIdx.x) >> 5);
  constexpr int NTL = (OUTC * 4) / 32;          // 32
  constexpr int MT = (B * FD) / 32;             // 512
  if (wave >= MT * NTL) return;
  const int tm = wave / NTL, tn = wave - tm * NTL;
  const int m0 = tm * 32;
  const int n  = m0 >> 10;                      // sample (FD = 1024)
  const int pb = m0 & (FD - 1);                 // pixel base within 32x32 plane
  const float* A = ws + OFF_NEW + (size_t)n * INC * FD;   // A[row][c] = new[n][c][pb+row]
  const int n0 = tn * 32;
  v8f acc[2][2] = {{vzero(), vzero()}, {vzero(), vzero()}};
  gemm_tile<2, 2>(INC,
      [&](int m, int k) { return A[(size_t)k * FD + pb + m]; },
      [&](int k, int nn) { return up_w[(size_t)k * (OUTC * 4) + n0 + nn]; }, acc);
  float* g = ws + OFF_G;
  store_block<2, 2>(acc, [&](int r, int c, float v) {
      const int ncol = n0 + c;
      const int o = ncol >> 2, a = (ncol >> 1) & 1, bb = ncol & 1;
      const int p = pb + r;
      const int yy = 2 * (p >> 5) + a, xx = 2 * (p & 31) + bb;
      const size_t sidx = (((size_t)n * OUTC + o) << 12) + (size_t)yy * W2 + xx;
      const float u = v + up_b[o] + s[sidx];
      const float rl = fmaxf(u, 0.f);
      g[sidx] = 1.f / (1.f + __expf(-rl));
  });
}

// ---------------- K6b: transpose conv weights -> w2t[tap][ic][oc] ----------------
__global__ void k_wt(const float* __restrict__ c1_w, float* __restrict__ ws) {
  const int i = blockIdx.x * blockDim.x + threadIdx.x;
  if (i >= 9 * IC2 * OUTC) return;
  const int oc = i & (OUTC - 1);
  const int r  = i >> 8;               // tap*IC2 + ic
  const int ic = r & (IC2 - 1);
  const int tap = r >> 9;
  ws[OFF_W2 + i] = c1_w[(size_t)oc * (IC2 * 9) + (size_t)ic * 9 + tap];
}

// ---------------- K7: Conv3x3 'SAME' as 9-tap implicit GEMM, TDM-staged weights -----
// block: 256 thr = 8 waves; each wave a 32(pixel) x 32(oc) C block; all waves share oc range
// grid.x = (B*P2)/256 M-blocks, grid.y = OUTC/32 N-blocks
constexpr int CONV_KC = 256;            // k-chunk staged in LDS: 256 x 32 floats = 32 KB
__global__ __launch_bounds__(256) void k_conv3(const float* __restrict__ s,
                                               const float* __restrict__ c1_b,
                                               float* __restrict__ ws) {
  __shared__ float bsh[CONV_KC * 32];
  const int n0 = blockIdx.y * 32;
  const int wave = threadIdx.x >> 5;
  const int m0 = blockIdx.x * 256 + wave * 32;  // 32 consecutive x in one image row
  const int n = m0 >> 12;                       // sample (P2 = 4096)
  const int yrow = (m0 & (P2 - 1)) >> 6;
  const int x0 = m0 & 63;
  const float* gbuf = ws + OFF_G + (size_t)n * OUTC * P2;
  const float* sbuf = s + (size_t)n * OUTC * P2;
  const float* w2t = ws + OFF_W2;
  const unsigned bsh_lds = (unsigned)(uintptr_t)&bsh[0];  // LDS aperture: addr[31:0] = LDS byte offset
  v8f acc[2][2] = {{vzero(), vzero()}, {vzero(), vzero()}};
#pragma unroll
  for (int tap = 0; tap < 9; ++tap) {
    const int dy = tap / 3 - 1, dx = tap % 3 - 1;
    const int yy = yrow + dy;
    const bool yok = ((unsigned)yy < (unsigned)H2);
    for (int kc = 0; kc < IC2; kc += CONV_KC) {
      __syncthreads();                          // previous chunk fully consumed
      if (wave == 0) {
        // TDM DMA: weight chunk [kc..kc+KC) x [n0..n0+32) -> LDS, no VGPR traffic
        const float* wsrc = w2t + ((size_t)tap * IC2 + kc) * OUTC + n0;
        tdm_load_2d_f32(bsh_lds, wsrc,
                        /*tensor_d0=*/OUTC, /*tensor_d1=*/9 * IC2,
                        /*stride_d0=*/OUTC,
                        /*tile_d0=*/32, /*tile_d1=*/CONV_KC);
        __builtin_amdgcn_s_wait_tensorcnt(0);
      }
      __syncthreads();                          // publish LDS tile to all 8 waves
      // A source: channels [kc..kc+KC) come entirely from g (kc=0) or s (kc=256)
      const float* abase = (kc == 0) ? gbuf : sbuf;
      gemm_tile<2, 2>(CONV_KC,
          [&](int m, int k) {
              const int xx = x0 + m + dx;
              if (!yok || (unsigned)xx >= (unsigned)W2) return 0.f;
              return abase[(size_t)k * P2 + (yy << 6) + xx];
          },
          [&](int k, int nn) { return bsh[k * 32 + nn]; }, acc);
    }
  }
  float* y = ws + OFF_Y;
  store_block<2, 2>(acc, [&](int r, int c, float v) {
      const int oc = n0 + c;
      y[(((size_t)n * OUTC + oc) << 12) + (yrow << 6) + x0 + r] = v + c1_b[oc];
  });
}

// ---------------- K8: per-channel batch stats (LDS tree reduction) ----------------
__global__ void k_bnstats(float* __restrict__ ws) {
  const int oc = blockIdx.x;                    // 256 blocks
  const float* y = ws + OFF_Y;
  float s = 0.f, s2 = 0.f;
  for (int i = threadIdx.x; i < B * P2; i += blockDim.x) {
    const int n = i >> 12, p = i & (P2 - 1);
    const float v = y[(((size_t)n * OUTC + oc) << 12) + p];
    s += v; s2 += v * v;
  }
  __shared__ float sh[512];
  sh[threadIdx.x] = s;
  sh[256 + threadIdx.x] = s2;
  __syncthreads();
  for (int off = 128; off > 0; off >>= 1) {
    if ((int)threadIdx.x < off) {
      sh[threadIdx.x] += sh[threadIdx.x + off];
      sh[256 + threadIdx.x] += sh[256 + threadIdx.x + off];
    }
    __syncthreads();
  }
  if (threadIdx.x == 0) {
    const float inv = 1.f / (float)(B * P2);
    const float mu = sh[0] * inv;
    const float var = sh[256] * inv - mu * mu;
    ws[OFF_ST + oc] = mu;
    ws[OFF_ST + OUTC + oc] = rsqrtf(var + 1e-5f);
  }
}

// ---------------- K9: BN apply + ReLU -> d_out ----------------
__global__ void k_bnapply(const float* __restrict__ bn_g,
                          const float* __restrict__ bn_b,
                          const float* __restrict__ ws,
                          float* __restrict__ out) {
  const size_t total4 = (size_t)B * OUTC * P2 / 4;
  size_t i4 = (size_t)blockIdx.x * blockDim.x + threadIdx.x;
  if (i4 >= total4) return;
  const size_t i = i4 * 4;
  const int oc = (int)((i >> 12) & (OUTC - 1));
  const float rstd = ws[OFF_ST + OUTC + oc];
  const float ga = bn_g[oc] * rstd;
  const float be = bn_b[oc] - ws[OFF_ST + oc] * ga;
  const float4 v = *(const float4*)(ws + OFF_Y + i);
  float4 o;
  o.x = fmaxf(v.x * ga + be, 0.f);
  o.y = fmaxf(v.y * ga + be, 0.f);
  o.z = fmaxf(v.z * ga + be, 0.f);
  o.w = fmaxf(v.w * ga + be, 0.f);
  *(float4*)(out + i) = o;
}

// ---------------- driver ----------------
extern "C" void kernel_launch(void* const* d_in, const int* in_sizes, int n_in,
                              void* d_out, int out_size, void* d_ws, size_t ws_size,
                              hipStream_t stream) {
  (void)in_sizes; (void)n_in; (void)out_size; (void)ws_size;
  const float* x    = (const float*)d_in[0];
  const float* s    = (const float*)d_in[1];
  const float* up_w = (const float*)d_in[2];
  const float* up_b = (const float*)d_in[3];
  const float* wq   = (const float*)d_in[4];
  const float* wkl  = (const float*)d_in[5];
  const float* wkr  = (const float*)d_in[6];
  const float* c1_w = (const float*)d_in[7];
  const float* c1_b = (const float*)d_in[8];
  const float* bn_g = (const float*)d_in[9];
  const float* bn_b = (const float*)d_in[10];
  float* ws  = (float*)d_ws;
  float* out = (float*)d_out;

  constexpr int THREADS = 256;                 // 8 wave32 per block
  constexpr int WPB = THREADS / 32;

  // stage 0: pass-through edge samples + conv weight transpose (independent)
  k_copy_edges<<<(INC * FD / 4 + THREADS - 1) / THREADS, THREADS, 0, stream>>>(x, ws);
  k_wt<<<(9 * IC2 * OUTC + THREADS - 1) / THREADS, THREADS, 0, stream>>>(c1_w, ws);

  // stage 1: attention
  {
    const int waves = 3 * NT * (INC / 32) * (DK / 32);
    k_proj<<<(waves + WPB - 1) / WPB, THREADS, 0, stream>>>(x, wq, wkl, wkr, ws);
  }
  k_ksum<<<(NT * 2 * DK + THREADS - 1) / THREADS, THREADS, 0, stream>>>(ws);
  {
    const int waves = NT * 2 * (DK / 32) * (FD / 32);
    k_kv<<<(waves + WPB - 1) / WPB, THREADS, 0, stream>>>(x, ws);
  }
  k_z<<<(NT * 2 * INC + THREADS - 1) / THREADS, THREADS, 0, stream>>>(ws);
  {
    const int waves = NT * (INC / 32) * (FD / 32);
    k_attn<<<(waves + WPB - 1) / WPB, THREADS, 0, stream>>>(x, ws);
  }

  // stage 2: transposed conv + gate
  {
    const int waves = (B * FD / 32) * (OUTC * 4 / 32);
    k_upgate<<<(waves + WPB - 1) / WPB, THREADS, 0, stream>>>(s, up_w, up_b, ws);
  }

  // stage 3: conv3x3 (TDM-staged weights in LDS)
  {
    dim3 grid((B * P2) / 256, OUTC / 32);
    k_conv3<<<grid, THREADS, 0, stream>>>(s, c1_b, ws);
  }

  // stage 4: batch-norm + relu
  k_bnstats<<<OUTC, THREADS, 0, stream>>>(ws);
  {
    const size_t total4 = (size_t)B * OUTC * P2 / 4;
    k_bnapply<<<(unsigned)((total4 + THREADS - 1) / THREADS), THREADS, 0, stream>>>(bn_g, bn_b, ws, out);
  }
}